// GCN_43456479101654
// MI455X (gfx1250) — compile-verified
//
#include <hip/hip_runtime.h>
#include <hip/hip_bf16.h>

typedef __attribute__((ext_vector_type(2))) float v2f;
typedef __attribute__((ext_vector_type(8))) float v8f;

// ---------------------------------------------------------------------------
// fp32 WMMA: D(16x16,f32) = A(16x4,f32) * B(4x16,f32) + C
// 8-arg VOP3P form: (neg_a, A, neg_b, B, c_mod, C, reuse_a, reuse_b)
// ---------------------------------------------------------------------------
__device__ __forceinline__ v8f wmma_f32_4(v2f a, v2f b, v8f c) {
    return __builtin_amdgcn_wmma_f32_16x16x4_f32(false, a, false, b,
                                                 (short)0, c, false, false);
}

// ---------------------------------------------------------------------------
// Degree / normalization kernels
// ---------------------------------------------------------------------------
__global__ void deg_init_kernel(float* deg, int n) {
    int i = blockIdx.x * blockDim.x + threadIdx.x;
    if (i < n) deg[i] = 1.0f;  // self-loop contribution
}

__global__ void deg_accum_kernel(const long long* __restrict__ dst, float* deg,
                                 long long E) {
    long long e = (long long)blockIdx.x * blockDim.x + threadIdx.x;
    if (e < E) atomicAdd(&deg[dst[e]], 1.0f);
}

__global__ void deg_rsqrt_kernel(float* deg, int n) {
    int i = blockIdx.x * blockDim.x + threadIdx.x;
    if (i < n) deg[i] = rsqrtf(deg[i]);  // deg >= 1 always (self-loop)
}

// ---------------------------------------------------------------------------
// WMMA GEMM: H[N x KOUT] = X[N x KIN] @ W[KIN x KOUT]
// KIN=128, KOUT in {128, 64}. One wave per 16-row tile; the wave carries
// KOUT/16 f32 accumulators (v8f each) and sweeps K in steps of 4.
// W is staged in LDS once per workgroup (64 KB / 32 KB << 320 KB per WGP).
//
// Fragment layouts per CDNA5 ISA 7.12.2 (wave32):
//   A 16x4 : lane l -> m = l&15, k-pair = 2*(l>>4); VGPR0=K, VGPR1=K+1
//   B 4x16 : lane l -> n = l&15, k-pair = 2*(l>>4); VGPR0=K, VGPR1=K+1
//   C 16x16: VGPR v -> row m = 8*(l>>4) + v, col n = l&15
// ---------------------------------------------------------------------------
template <int KIN, int KOUT>
__global__ void gemm_wmma_kernel(const float* __restrict__ X,
                                 const float* __restrict__ W,
                                 float* __restrict__ H, int nTiles) {
    __shared__ float wlds[KIN * KOUT];
    for (int i = threadIdx.x; i < KIN * KOUT; i += blockDim.x)
        wlds[i] = W[i];
    __syncthreads();

    const int lane = threadIdx.x & 31;
    const int wave = threadIdx.x >> 5;
    const int tile = blockIdx.x * (blockDim.x >> 5) + wave;
    if (tile >= nTiles) return;  // wave-uniform: EXEC stays all-1s for WMMA

    const int mn = lane & 15;          // m for A, n for B/C
    const int kk = (lane >> 4) << 1;   // which K-pair this half-wave holds

    const float* __restrict__ xrow = X + (size_t)(tile * 16 + mn) * KIN;

    constexpr int NT = KOUT / 16;
    v8f acc[NT];
#pragma unroll
    for (int t = 0; t < NT; ++t)
#pragma unroll
        for (int v = 0; v < 8; ++v) acc[t][v] = 0.0f;

    for (int k = 0; k < KIN; k += 4) {
        v2f a;
        a.x = xrow[k + kk];
        a.y = xrow[k + kk + 1];
        const float* wr0 = &wlds[(k + kk) * KOUT];
        const float* wr1 = &wlds[(k + kk + 1) * KOUT];
#pragma unroll
        for (int t = 0; t < NT; ++t) {
            const int n = t * 16 + mn;
            v2f b;
            b.x = wr0[n];
            b.y = wr1[n];
            acc[t] = wmma_f32_4(a, b, acc[t]);
        }
    }

    const int mbase = (lane >> 4) * 8;
    float* __restrict__ hrow = H + (size_t)(tile * 16 + mbase) * KOUT;
#pragma unroll
    for (int t = 0; t < NT; ++t)
#pragma unroll
        for (int v = 0; v < 8; ++v)
            hrow[(size_t)v * KOUT + t * 16 + mn] = acc[t][v];
}

// ---------------------------------------------------------------------------
// acc[i] = h[i] * dinv[node]^2   (self-loop message initializes accumulator)
// ---------------------------------------------------------------------------
template <int F>
__global__ void self_init_kernel(const float* __restrict__ h,
                                 const float* __restrict__ dinv,
                                 float* __restrict__ acc, int n) {
    long long i = (long long)blockIdx.x * blockDim.x + threadIdx.x;
    if (i < (long long)n * F) {
        const int node = (int)(i / F);
        const float dv = dinv[node];
        acc[i] = h[i] * dv * dv;
    }
}

// ---------------------------------------------------------------------------
// Edge scatter: acc[dst[e]][f] += h[src[e]][f] * dinv[src]*dinv[dst]
// One thread per (edge, feature). h/acc are L2-resident (51 MB << 192 MB L2)
// so gathers and fp32 atomics resolve in L2; HBM only streams the edge list.
// ---------------------------------------------------------------------------
template <int F>
__global__ void scatter_kernel(const float* __restrict__ h,
                               const float* __restrict__ dinv,
                               const long long* __restrict__ src,
                               const long long* __restrict__ dst,
                               float* __restrict__ acc, long long E) {
    long long idx = (long long)blockIdx.x * blockDim.x + threadIdx.x;
    if (idx >= E * F) return;
    const long long e = idx / F;
    const int f = (int)(idx % F);
    const long long s = src[e];
    const long long d = dst[e];
    const float nrm = dinv[s] * dinv[d];
    atomicAdd(&acc[d * F + f], h[s * F + f] * nrm);
}

// ---------------------------------------------------------------------------
// In-place bias + ReLU over [n x F]
// ---------------------------------------------------------------------------
template <int F>
__global__ void bias_relu_kernel(float* __restrict__ a,
                                 const float* __restrict__ b, int n) {
    long long i = (long long)blockIdx.x * blockDim.x + threadIdx.x;
    if (i < (long long)n * F) {
        const float v = a[i] + b[(int)(i % F)];
        a[i] = v > 0.0f ? v : 0.0f;
    }
}

// ---------------------------------------------------------------------------
// out = log_softmax(acc + b) per row of 64; one wave32 per row, 2 vals/lane,
// shuffle reductions (wave32 semantics on gfx1250).
// ---------------------------------------------------------------------------
__global__ void bias_logsoftmax_kernel(const float* __restrict__ acc,
                                       const float* __restrict__ b,
                                       float* __restrict__ out, int n) {
    const int lane = threadIdx.x & 31;
    const int wave = threadIdx.x >> 5;
    const int row = blockIdx.x * (blockDim.x >> 5) + wave;
    if (row >= n) return;

    const float* r = acc + (size_t)row * 64;
    float v0 = r[lane] + b[lane];
    float v1 = r[lane + 32] + b[lane + 32];

    float m = fmaxf(v0, v1);
#pragma unroll
    for (int o = 16; o > 0; o >>= 1) m = fmaxf(m, __shfl_xor(m, o, 32));

    float s = __expf(v0 - m) + __expf(v1 - m);
#pragma unroll
    for (int o = 16; o > 0; o >>= 1) s += __shfl_xor(s, o, 32);

    const float L = m + __logf(s);
    out[(size_t)row * 64 + lane] = v0 - L;
    out[(size_t)row * 64 + lane + 32] = v1 - L;
}

// ---------------------------------------------------------------------------
// Host launch
// Inputs: x[N*128], edge_index[2*E] (int64), W1[128*128], b1[128],
//         W2[128*64], b2[64].  Output: [N*64] f32 log-probs.
// ---------------------------------------------------------------------------
extern "C" void kernel_launch(void* const* d_in, const int* in_sizes, int n_in,
                              void* d_out, int out_size, void* d_ws, size_t ws_size,
                              hipStream_t stream) {
    const float* x = (const float*)d_in[0];
    const long long* edge = (const long long*)d_in[1];  // int64 per reference
    const float* W1 = (const float*)d_in[2];
    const float* b1 = (const float*)d_in[3];
    const float* W2 = (const float*)d_in[4];
    const float* b2 = (const float*)d_in[5];
    float* out = (float*)d_out;

    const int DIN = 128, DHID = 128, DOUT = 64;
    const int N = in_sizes[0] / DIN;
    const long long E = in_sizes[1] / 2;
    const long long* src = edge;
    const long long* dst = edge + E;

    // workspace layout (256B aligned regions)
    auto alignup = [](size_t x) { return (x + 255) & ~(size_t)255; };
    char* w = (char*)d_ws;
    float* dinv = (float*)w; w += alignup((size_t)N * sizeof(float));
    float* h1   = (float*)w; w += alignup((size_t)N * DHID * sizeof(float));
    float* acc1 = (float*)w; w += alignup((size_t)N * DHID * sizeof(float));
    float* h2   = (float*)w; w += alignup((size_t)N * DOUT * sizeof(float));
    float* acc2 = (float*)w;

    const int nTiles = (N + 15) / 16;       // N=100000 -> 6250 exact
    const int wavesPerBlk = 8;              // 256 threads
    const int gemmBlocks = (nTiles + wavesPerBlk - 1) / wavesPerBlk;

    // 1) symmetric normalization: dinv = rsqrt(1 + in-degree)
    deg_init_kernel<<<(N + 255) / 256, 256, 0, stream>>>(dinv, N);
    deg_accum_kernel<<<(int)((E + 255) / 256), 256, 0, stream>>>(dst, dinv, E);
    deg_rsqrt_kernel<<<(N + 255) / 256, 256, 0, stream>>>(dinv, N);

    // 2) layer 1: h1 = x @ W1  (WMMA fp32)
    gemm_wmma_kernel<128, 128><<<gemmBlocks, 256, 0, stream>>>(x, W1, h1, nTiles);

    // 3) aggregate: acc1 = self + scatter
    {
        long long tot = (long long)N * DHID;
        self_init_kernel<128><<<(int)((tot + 255) / 256), 256, 0, stream>>>(h1, dinv, acc1, N);
        long long work = E * DHID;
        scatter_kernel<128><<<(int)((work + 255) / 256), 256, 0, stream>>>(h1, dinv, src, dst, acc1, E);
        bias_relu_kernel<128><<<(int)((tot + 255) / 256), 256, 0, stream>>>(acc1, b1, N);
    }

    // 4) layer 2: h2 = relu(acc1) @ W2  (WMMA fp32)
    gemm_wmma_kernel<128, 64><<<gemmBlocks, 256, 0, stream>>>(acc1, W2, h2, nTiles);

    // 5) aggregate layer 2
    {
        long long tot = (long long)N * DOUT;
        self_init_kernel<64><<<(int)((tot + 255) / 256), 256, 0, stream>>>(h2, dinv, acc2, N);
        long long work = E * DOUT;
        scatter_kernel<64><<<(int)((work + 255) / 256), 256, 0, stream>>>(h2, dinv, src, dst, acc2, E);
    }

    // 6) bias + log_softmax -> out
    bias_logsoftmax_kernel<<<(N + wavesPerBlk - 1) / wavesPerBlk, 256, 0, stream>>>(acc2, b2, out, N);
}